// SNNLoss_41609643164116
// MI455X (gfx1250) — compile-verified
//
#include <hip/hip_runtime.h>
#include <hip/hip_bf16.h>
#include <math.h>

typedef __attribute__((ext_vector_type(2))) float v2f;
typedef __attribute__((ext_vector_type(8))) float v8f;

#define B_ROWS 8192
#define DIMS 512
#define PAD 516            // DIMS + 4 floats: conflict-free ds_load_b64 fragments
#define ROWS_PER_BLOCK 128
#define COLS_PER_STEP 16
#define NUM_COL_STEPS (B_ROWS / COLS_PER_STEP)

// Fast hardware transcendentals (raw v_sqrt_f32 / v_exp_f32 / v_log_f32).
// Operand ranges here make them safe: sqd is O(1e3) (no denorms), exp2 args
// are <= 0 and underflow cleanly to 0, log2 args are finite positive.
__device__ __forceinline__ float fsqrt_hw(float x) { return __builtin_amdgcn_sqrtf(x); }
__device__ __forceinline__ float fexp2_hw(float x) { return __builtin_amdgcn_exp2f(x); }
__device__ __forceinline__ float flog2_hw(float x) { return __builtin_amdgcn_logf(x); }

// ws layout (floats):
// [0, 8192)       rowsum
// [8192, 16384)   rowsumsq (raw squared norms, reused as sqn)
// [16384, 24576)  rowval   (den - num per row, natural log units)
// [24576]         coef2 = inv_std * 10^T / ln2

// ---------------- Kernel 1: per-row sum & sumsq (raw x) ----------------
__global__ __launch_bounds__(128) void rowstats_kernel(
    const float* __restrict__ x, float* __restrict__ rowsum, float* __restrict__ rowsumsq) {
  int row = blockIdx.x;
  int tid = threadIdx.x;                      // 128 threads, 1 float4 each
  float4 v = ((const float4*)(x + (size_t)row * DIMS))[tid];
  float s = v.x + v.y + v.z + v.w;
  float q = v.x * v.x + v.y * v.y + v.z * v.z + v.w * v.w;
  for (int off = 16; off >= 1; off >>= 1) {
    s += __shfl_xor(s, off, 32);
    q += __shfl_xor(q, off, 32);
  }
  __shared__ float ss[4], qq[4];
  int w = tid >> 5;
  if ((tid & 31) == 0) { ss[w] = s; qq[w] = q; }
  __syncthreads();
  if (tid == 0) {
    rowsum[row]   = ss[0] + ss[1] + ss[2] + ss[3];
    rowsumsq[row] = qq[0] + qq[1] + qq[2] + qq[3];
  }
}

// ---------------- Kernel 2: global std (ddof=1) + temperature ----------------
__global__ __launch_bounds__(256) void stats_finalize_kernel(
    const float* __restrict__ rowsum, const float* __restrict__ rowsumsq,
    const float* __restrict__ Tptr, float* __restrict__ params) {
  __shared__ float ssum[256], ssq[256];
  int tid = threadIdx.x;
  float s = 0.f, q = 0.f;
  for (int i = tid; i < B_ROWS; i += 256) { s += rowsum[i]; q += rowsumsq[i]; }
  ssum[tid] = s; ssq[tid] = q;
  __syncthreads();
  for (int st = 128; st >= 1; st >>= 1) {
    if (tid < st) { ssum[tid] += ssum[tid + st]; ssq[tid] += ssq[tid + st]; }
    __syncthreads();
  }
  if (tid == 0) {
    double n   = (double)B_ROWS * (double)DIMS;
    double sum = (double)ssum[0], sq = (double)ssq[0];
    double var = (sq - sum * sum / n) / (n - 1.0);   // unbiased, like torch.std
    float inv_std = (float)(1.0 / sqrt(var));
    float T   = Tptr[0];
    float mul = exp2f(T * 3.3219280948873623f);      // 10^T
    params[0] = inv_std * mul * 1.4426950408889634f; // fold 1/ln2: work in base-2
  }
}

// ---------------- Kernel 3: fused WMMA Gram + online logsumexp ----------------
__global__ __launch_bounds__(256) void snn_main_kernel(
    const float* __restrict__ x, const long long* __restrict__ y,
    const float* __restrict__ sqn_raw, const float* __restrict__ params,
    float* __restrict__ rowval) {
  extern __shared__ float smem[];
  float* As = smem;                          // [128][PAD]
  float* Bs = smem + ROWS_PER_BLOCK * PAD;   // [16][PAD]

  const int tid  = threadIdx.x;
  const int wave = tid >> 5;
  const int lane = tid & 31;
  const int half = lane >> 4;
  const int l15  = lane & 15;

  const int rowBlock = blockIdx.x * ROWS_PER_BLOCK;
  const int waveBase = rowBlock + wave * 16;
  const int rbase    = waveBase + half * 8;  // this lane's 8 contiguous rows (C layout)

  const float coef2 = params[0];

  // Stage A block (128 rows x 512 f32) into padded LDS, float4 vectorized
  for (int i = 0; i < 64; ++i) {
    int linear = tid + i * 256;              // 16384 float4 total
    int row = linear >> 7;                   // /128 float4-per-row
    int kq  = linear & 127;
    float4 v = ((const float4*)(x + (size_t)(rowBlock + row) * DIMS))[kq];
    *(float4*)&As[row * PAD + kq * 4] = v;
  }

  // Per-lane row metadata
  float sqn_i[8]; int ylab[8];
  #pragma unroll
  for (int r = 0; r < 8; ++r) {
    sqn_i[r] = sqn_raw[rbase + r];
    ylab[r]  = (int)y[rbase + r];
  }

  float mden[8], sden[8], mnum[8], snum[8];
  #pragma unroll
  for (int r = 0; r < 8; ++r) { mden[r] = -1e30f; sden[r] = 0.f; mnum[r] = -1e30f; snum[r] = 0.f; }

  // WMMA fragment base pointers (ISA A 16x4 f32 layout; B mirrors since B = rows of x)
  const float* aptr = &As[(wave * 16 + l15) * PAD + 2 * half];
  const float* bptr = &Bs[l15 * PAD + 2 * half];

  for (int cb = 0; cb < NUM_COL_STEPS; ++cb) {
    const int colBase = cb * COLS_PER_STEP;
    __syncthreads();
    // Stage B tile (16 cols' rows x 512)
    for (int i = 0; i < 8; ++i) {
      int linear = tid + i * 256;            // 2048 float4
      int col = linear >> 7;
      int kq  = linear & 127;
      float4 v = ((const float4*)(x + (size_t)(colBase + col) * DIMS))[kq];
      *(float4*)&Bs[col * PAD + kq * 4] = v;
    }
    __syncthreads();

    // 16x16 f32 Gram tile, K = 512 via 128 x v_wmma_f32_16x16x4_f32
    v8f c = {0.f, 0.f, 0.f, 0.f, 0.f, 0.f, 0.f, 0.f};
    #pragma unroll 8
    for (int k = 0; k < DIMS; k += 4) {
      v2f a = *(const v2f*)(aptr + k);
      v2f b = *(const v2f*)(bptr + k);
      c = __builtin_amdgcn_wmma_f32_16x16x4_f32(false, a, false, b, (short)0, c,
                                                false, false);
    }

    const int   gj    = colBase + l15;
    const float sqn_j = sqn_raw[gj];
    const int   yj    = (int)y[gj];

    // Online base-2 logsumexp update (flash-style), per lane column subset
    #pragma unroll
    for (int r = 0; r < 8; ++r) {
      const int gi = rbase + r;
      float sqd = fmaxf(sqn_i[r] + sqn_j - 2.0f * c[r], 0.0f);   // raw, scale folded in coef2
      float e2  = -fsqrt_hw(sqd) * coef2;                        // e_dist / ln2
      if (gi != gj) {
        float nm = fmaxf(mden[r], e2);
        sden[r] = sden[r] * fexp2_hw(mden[r] - nm) + fexp2_hw(e2 - nm);
        mden[r] = nm;
        if (ylab[r] == yj) {
          float nm2 = fmaxf(mnum[r], e2);
          snum[r] = snum[r] * fexp2_hw(mnum[r] - nm2) + fexp2_hw(e2 - nm2);
          mnum[r] = nm2;
        }
      }
    }
  }

  // Combine partial (m, s) across the 16 lanes of each half-wave
  #pragma unroll
  for (int r = 0; r < 8; ++r) {
    float m = mden[r], s = sden[r];
    float mn = mnum[r], sn = snum[r];
    #pragma unroll
    for (int off = 8; off >= 1; off >>= 1) {
      float mo = __shfl_xor(m, off, 32);
      float so = __shfl_xor(s, off, 32);
      float nm = fmaxf(m, mo);
      s = s * fexp2_hw(m - nm) + so * fexp2_hw(mo - nm);
      m = nm;
      float mno = __shfl_xor(mn, off, 32);
      float sno = __shfl_xor(sn, off, 32);
      float nmn = fmaxf(mn, mno);
      sn = sn * fexp2_hw(mn - nmn) + sno * fexp2_hw(mno - nmn);
      mn = nmn;
    }
    if (l15 == 0) {
      float den_l2 = m + flog2_hw(s);                 // den always has 8191 terms
      float val = 0.0f;                               // rows w/o positives contribute 0
      if (sn > 0.0f) {
        float num_l2 = mn + flog2_hw(sn);
        val = (den_l2 - num_l2) * 0.6931471805599453f; // back to natural log
      }
      rowval[rbase + r] = val;
    }
  }
}

// ---------------- Kernel 4: deterministic mean ----------------
__global__ __launch_bounds__(256) void final_mean_kernel(
    const float* __restrict__ rowval, float* __restrict__ out) {
  __shared__ float sm[256];
  int tid = threadIdx.x;
  float s = 0.f;
  for (int i = tid; i < B_ROWS; i += 256) s += rowval[i];
  sm[tid] = s;
  __syncthreads();
  for (int st = 128; st >= 1; st >>= 1) {
    if (tid < st) sm[tid] += sm[tid + st];
    __syncthreads();
  }
  if (tid == 0) out[0] = sm[0] * (1.0f / (float)B_ROWS);  // loss = mean(den - num)
}

extern "C" void kernel_launch(void* const* d_in, const int* in_sizes, int n_in,
                              void* d_out, int out_size, void* d_ws, size_t ws_size,
                              hipStream_t stream) {
  const float*     x = (const float*)d_in[0];
  const long long* y = (const long long*)d_in[1];
  const float*     T = (const float*)d_in[2];
  float* ws       = (float*)d_ws;
  float* rowsum   = ws;
  float* rowsumsq = ws + B_ROWS;
  float* rowval   = ws + 2 * B_ROWS;
  float* params   = ws + 3 * B_ROWS;

  rowstats_kernel<<<B_ROWS, 128, 0, stream>>>(x, rowsum, rowsumsq);
  stats_finalize_kernel<<<1, 256, 0, stream>>>(rowsum, rowsumsq, T, params);

  size_t shbytes = (size_t)(ROWS_PER_BLOCK + COLS_PER_STEP) * PAD * sizeof(float);
  snn_main_kernel<<<B_ROWS / ROWS_PER_BLOCK, 256, shbytes, stream>>>(
      x, y, rowsumsq, params, rowval);

  final_mean_kernel<<<1, 256, 0, stream>>>(rowval, (float*)d_out);
}